// GGNNSum_19645180412195
// MI455X (gfx1250) — compile-verified
//
#include <hip/hip_runtime.h>
#include <math.h>

// ---------------------------------------------------------------------------
// GGNN (GatedGraphConv, 8 steps) for gfx1250 / MI455X.
// fp32 WMMA (V_WMMA_F32_16X16X4_F32) for all GEMMs; edge phase is the
// bandwidth-bound part and uses contiguous float4 reads + hw f32 atomics.
// ---------------------------------------------------------------------------

#define N_NODES  32768
#define N_GRAPHS 64
#define N_EDGES  524288
#define D        128      // D_IN == D_OUT == 128 (so the zero-pad is a no-op)
#define N_ETYPES 4
#define N_STEPS  8

typedef float v2f __attribute__((ext_vector_type(2)));
typedef float v8f __attribute__((ext_vector_type(8)));

__device__ __forceinline__ v8f wmma_f32(v2f a, v2f b, v8f c) {
    // 8 args: (neg_a, A, neg_b, B, c_mod, C, reuse_a, reuse_b)
    return __builtin_amdgcn_wmma_f32_16x16x4_f32(false, a, false, b,
                                                 (short)0, c, false, false);
}

__device__ __forceinline__ float sigmoidf_(float x) {
    return 1.0f / (1.0f + expf(-x));
}

// ---------------------------------------------------------------------------
// Kernel 1: tm[t][n][k] = sum_d h[n][d] * W_edge[t][d][k] + b_edge[t][k]
// One wave computes a 16-node x 128-feature strip for one edge type.
// grid = (N_NODES/16/8, N_ETYPES), block = 256 (8 waves).
// ---------------------------------------------------------------------------
__global__ void edge_transform_kernel(const float* __restrict__ h,
                                      const float* __restrict__ W_edge,
                                      const float* __restrict__ b_edge,
                                      float* __restrict__ tm) {
    const int lane = threadIdx.x & 31;
    const int wv   = threadIdx.x >> 5;
    const int t    = blockIdx.y;
    const int n0   = (blockIdx.x * 8 + wv) * 16;

    const float* __restrict__ W    = W_edge + (size_t)t * D * D;  // [d][k]
    const float* __restrict__ bias = b_edge + (size_t)t * D;
    float* __restrict__ out        = tm + ((size_t)t * N_NODES + n0) * D;

    const int row  = lane & 15;         // M index (A) / N index (B,C)
    const int koff = (lane >> 4) * 2;   // K pair selector

    v8f zero = {0.f, 0.f, 0.f, 0.f, 0.f, 0.f, 0.f, 0.f};
    v8f acc[8];
#pragma unroll
    for (int j = 0; j < 8; ++j) acc[j] = zero;

    const float* hrow = h + (size_t)(n0 + row) * D;
    for (int k0 = 0; k0 < D; k0 += 4) {
        v2f a;
        a.x = hrow[k0 + koff];
        a.y = hrow[k0 + koff + 1];
#pragma unroll
        for (int j = 0; j < 8; ++j) {
            const int col = j * 16 + row;
            v2f b;
            b.x = W[(size_t)(k0 + koff) * D + col];
            b.y = W[(size_t)(k0 + koff + 1) * D + col];
            acc[j] = wmma_f32(a, b, acc[j]);
        }
    }

    const int rbase = (lane >> 4) * 8;  // C/D: VGPR v -> M = v + 8*(lane>>4)
#pragma unroll
    for (int j = 0; j < 8; ++j) {
        const int col = j * 16 + row;
        const float bk = bias[col];
#pragma unroll
        for (int v = 0; v < 8; ++v) {
            const int r = rbase + v;
            out[(size_t)r * D + col] = acc[j][v] + bk;
        }
    }
}

// ---------------------------------------------------------------------------
// Kernel 2: per-edge gather + scatter-add.
// One wave per edge: 512B contiguous read of tm[etype][src], hw f32 atomics
// into a[dst]. grid = N_EDGES*32/256 blocks.
// ---------------------------------------------------------------------------
__global__ void edge_message_kernel(const float* __restrict__ tm,
                                    const int* __restrict__ src,
                                    const int* __restrict__ dst,
                                    const int* __restrict__ ety,
                                    float* __restrict__ a) {
    const int gid  = blockIdx.x * blockDim.x + threadIdx.x;
    const int e    = gid >> 5;
    const int lane = gid & 31;
    if (e >= N_EDGES) return;

    const int s = src[e];
    const int d = dst[e];
    const int t = ety[e];

    const float4* __restrict__ m =
        (const float4*)(tm + ((size_t)t * N_NODES + s) * D);
    float* __restrict__ outp = a + (size_t)d * D + lane * 4;

    const float4 v = m[lane];  // 32 lanes * 16B = 512B = full row
    unsafeAtomicAdd(outp + 0, v.x);
    unsafeAtomicAdd(outp + 1, v.y);
    unsafeAtomicAdd(outp + 2, v.z);
    unsafeAtomicAdd(outp + 3, v.w);
}

// ---------------------------------------------------------------------------
// Kernel 3: fused GRU cell.
//   gi = a @ w_ih^T + b_ih   (w_ih: [384][128] row-major -> B[k][j]=w_ih[j*128+k])
//   gh = h @ w_hh^T + b_hh
//   r = sig(gi_r+gh_r); z = sig(gi_z+gh_z); n = tanh(gi_n + r*gh_n)
//   h' = (1-z)*n + z*h
// Block = 256 threads (8 waves) handles 16 nodes. Wave w owns 48 output cols
// of BOTH gi and gh (3+3 accumulators). gi/gh staged in LDS for gate math.
// grid = N_NODES/16.
// ---------------------------------------------------------------------------
__global__ void gru_kernel(const float* __restrict__ a,
                           const float* __restrict__ h,
                           const float* __restrict__ w_ih,
                           const float* __restrict__ w_hh,
                           const float* __restrict__ b_ih,
                           const float* __restrict__ b_hh,
                           float* __restrict__ h_out) {
    __shared__ float gi_s[16 * 384];
    __shared__ float gh_s[16 * 384];

    const int tid  = threadIdx.x;
    const int wv   = tid >> 5;
    const int lane = tid & 31;
    const int n0   = blockIdx.x * 16;

    const int row   = lane & 15;
    const int koff  = (lane >> 4) * 2;
    const int jbase = wv * 48;

    v8f zero = {0.f, 0.f, 0.f, 0.f, 0.f, 0.f, 0.f, 0.f};
    v8f gi_acc[3], gh_acc[3];
#pragma unroll
    for (int j = 0; j < 3; ++j) { gi_acc[j] = zero; gh_acc[j] = zero; }

    const float* arow = a + (size_t)(n0 + row) * D;
    const float* hrow = h + (size_t)(n0 + row) * D;

    for (int k0 = 0; k0 < D; k0 += 4) {
        v2f af, hf;
        af.x = arow[k0 + koff];
        af.y = arow[k0 + koff + 1];
        hf.x = hrow[k0 + koff];
        hf.y = hrow[k0 + koff + 1];
#pragma unroll
        for (int jt = 0; jt < 3; ++jt) {
            const int col = jbase + jt * 16 + row;   // output gate column j
            v2f b1, b2;
            b1.x = w_ih[(size_t)col * D + k0 + koff];
            b1.y = w_ih[(size_t)col * D + k0 + koff + 1];
            b2.x = w_hh[(size_t)col * D + k0 + koff];
            b2.y = w_hh[(size_t)col * D + k0 + koff + 1];
            gi_acc[jt] = wmma_f32(af, b1, gi_acc[jt]);
            gh_acc[jt] = wmma_f32(hf, b2, gh_acc[jt]);
        }
    }

    const int rbase = (lane >> 4) * 8;
#pragma unroll
    for (int jt = 0; jt < 3; ++jt) {
        const int col = jbase + jt * 16 + row;
        const float bi = b_ih[col];
        const float bh = b_hh[col];
#pragma unroll
        for (int v = 0; v < 8; ++v) {
            const int r = rbase + v;
            gi_s[r * 384 + col] = gi_acc[jt][v] + bi;
            gh_s[r * 384 + col] = gh_acc[jt][v] + bh;
        }
    }
    __syncthreads();

    // gate math: 16 nodes * 128 feats = 2048 elems, 8 per thread
#pragma unroll
    for (int e = 0; e < 8; ++e) {
        const int idx  = tid + 256 * e;
        const int node = idx >> 7;
        const int d    = idx & 127;
        const float rr = sigmoidf_(gi_s[node * 384 + d] + gh_s[node * 384 + d]);
        const float zz = sigmoidf_(gi_s[node * 384 + 128 + d] + gh_s[node * 384 + 128 + d]);
        const float nn = tanhf(gi_s[node * 384 + 256 + d] + rr * gh_s[node * 384 + 256 + d]);
        const float hp = h[(size_t)(n0 + node) * D + d];
        h_out[(size_t)(n0 + node) * D + d] = (1.0f - zz) * nn + zz * hp;
    }
}

// ---------------------------------------------------------------------------
// Kernel 4: graph readout. logits[g] = sum_{n in g} h[n][:] . w_cls + b_cls
// One block per graph (512 nodes each).
// ---------------------------------------------------------------------------
__global__ void readout_kernel(const float* __restrict__ h,
                               const float* __restrict__ w_cls,
                               const float* __restrict__ b_cls,
                               float* __restrict__ out) {
    const int g = blockIdx.x;
    const float* __restrict__ hg = h + (size_t)g * 512 * D;

    float partial = 0.0f;
    for (int i = threadIdx.x; i < 512 * D; i += 256)
        partial += hg[i] * w_cls[i & (D - 1)];

    __shared__ float red[256];
    red[threadIdx.x] = partial;
    __syncthreads();
#pragma unroll
    for (int s = 128; s > 0; s >>= 1) {
        if (threadIdx.x < s) red[threadIdx.x] += red[threadIdx.x + s];
        __syncthreads();
    }
    if (threadIdx.x == 0)
        out[g] = sigmoidf_(red[0] + b_cls[0]);
}

// ---------------------------------------------------------------------------
extern "C" void kernel_launch(void* const* d_in, const int* in_sizes, int n_in,
                              void* d_out, int out_size, void* d_ws, size_t ws_size,
                              hipStream_t stream) {
    (void)in_sizes; (void)n_in; (void)out_size; (void)ws_size;

    const float* feat   = (const float*)d_in[0];
    const float* W_edge = (const float*)d_in[1];
    const float* b_edge = (const float*)d_in[2];
    const float* w_ih   = (const float*)d_in[3];
    const float* w_hh   = (const float*)d_in[4];
    const float* b_ih   = (const float*)d_in[5];
    const float* b_hh   = (const float*)d_in[6];
    const float* w_cls  = (const float*)d_in[7];
    const float* b_cls  = (const float*)d_in[8];
    const int*   src    = (const int*)d_in[9];
    const int*   dst    = (const int*)d_in[10];
    const int*   ety    = (const int*)d_in[11];
    float* out = (float*)d_out;

    // Workspace layout (fp32): hA | hB | tm[4] | a   ==> ~116 MB
    const size_t NF = (size_t)N_NODES * D;
    float* hA = (float*)d_ws;
    float* hB = hA + NF;
    float* tm = hB + NF;
    float* ag = tm + (size_t)N_ETYPES * NF;

    // h0 = zero-pad(feat) == feat since D_IN == D_OUT == 128
    hipMemcpyAsync(hA, feat, NF * sizeof(float), hipMemcpyDeviceToDevice, stream);

    float* hc = hA;
    float* hn = hB;
    for (int step = 0; step < N_STEPS; ++step) {
        edge_transform_kernel<<<dim3(N_NODES / 16 / 8, N_ETYPES), 256, 0, stream>>>(
            hc, W_edge, b_edge, tm);
        hipMemsetAsync(ag, 0, NF * sizeof(float), stream);
        edge_message_kernel<<<(N_EDGES * 32) / 256, 256, 0, stream>>>(
            tm, src, dst, ety, ag);
        gru_kernel<<<N_NODES / 16, 256, 0, stream>>>(
            ag, hc, w_ih, w_hh, b_ih, b_hh, hn);
        float* t_ = hc; hc = hn; hn = t_;
    }

    readout_kernel<<<N_GRAPHS, 256, 0, stream>>>(hc, w_cls, b_cls, out);
}